// NAAGCN_24481313587855
// MI455X (gfx1250) — compile-verified
//
#include <hip/hip_runtime.h>
#include <math.h>

typedef float v2f __attribute__((ext_vector_type(2)));
typedef float v8f __attribute__((ext_vector_type(8)));

#define IN_DIM 128
#define HID    64
#define OUT_D  2

// fp32 global atomic add without return -> global_atomic_add_f32 (STOREcnt path)
__device__ __forceinline__ void gatomic_add_f32(float* p, float v) {
  asm volatile("global_atomic_add_f32 %0, %1, off" :: "v"(p), "v"(v) : "memory");
}

// ---------------- small prep kernels ----------------

// W1p[k][j] = sigmoid(fi[k]) * W1[k][j]   (fold feature gating into W1)
__global__ void k_sigw1(const float* __restrict__ fi, const float* __restrict__ W1,
                        float* __restrict__ W1p) {
  int idx = blockIdx.x * blockDim.x + threadIdx.x;
  if (idx < IN_DIM * HID) {
    int k = idx / HID;
    float g = 1.0f / (1.0f + __expf(-fi[k]));
    W1p[idx] = W1[idx] * g;
  }
}

__global__ void k_fill1(float* __restrict__ deg, int n) {
  int i = blockIdx.x * blockDim.x + threadIdx.x;
  if (i < n) deg[i] = 1.0f;   // self-loop contribution
}

__global__ void k_degcount(const int* __restrict__ dst, float* __restrict__ deg, int e) {
  int i = blockIdx.x * blockDim.x + threadIdx.x;
  if (i < e) gatomic_add_f32(&deg[dst[i]], 1.0f);
}

__global__ void k_rsqrt(float* __restrict__ deg, int n) {
  int i = blockIdx.x * blockDim.x + threadIdx.x;
  if (i < n) deg[i] = rsqrtf(deg[i]);   // deg >= 1 always (self loops)
}

// ---------------- GEMM1: h0 = x @ W1p  via V_WMMA_F32_16X16X4_F32 ----------------
// One wave computes a 16x64 output tile. 4 waves / block. W1p staged in LDS.
__global__ void __launch_bounds__(128) k_gemm1(const float* __restrict__ x,
                                               const float* __restrict__ W1p,
                                               float* __restrict__ h0,
                                               int nNodes, int nTiles) {
  __shared__ float sB[IN_DIM * HID];   // 32 KB
  {
    const float4* s4 = (const float4*)W1p;
    float4* d4 = (float4*)sB;
    for (int i = threadIdx.x; i < (IN_DIM * HID) / 4; i += blockDim.x) d4[i] = s4[i];
  }
  __syncthreads();

  const int wave   = threadIdx.x >> 5;
  const int lane   = threadIdx.x & 31;
  const int tile   = blockIdx.x * 4 + wave;
  if (tile >= nTiles) return;          // wave-uniform: EXEC stays all-ones for WMMA

  const int lane16 = lane & 15;
  const int half   = lane >> 4;        // 0: K even pair base, 1: K odd pair base
  const int m0     = tile * 16;
  int rowA = m0 + lane16;
  if (rowA >= nNodes) rowA = nNodes - 1;   // clamp (n=100000 is a multiple of 16 anyway)
  const float* xrow = x + (long)rowA * IN_DIM;

  v8f c0 = {}, c1 = {}, c2 = {}, c3 = {};

#pragma unroll
  for (int k0 = 0; k0 < IN_DIM; k0 += 4) {
    const int kk = k0 + 2 * half;
    // A 16x4 f32 layout: VGPR0 = {K=kk}, VGPR1 = {K=kk+1}
    const float2 av = *(const float2*)(xrow + kk);
    v2f a; a.x = av.x; a.y = av.y;
    // B 4x16 f32 (mirrored layout): component0 = row kk, component1 = row kk+1
    const float* bp = sB + kk * HID + lane16;
    v2f b0; b0.x = bp[0];  b0.y = bp[HID + 0];
    v2f b1; b1.x = bp[16]; b1.y = bp[HID + 16];
    v2f b2; b2.x = bp[32]; b2.y = bp[HID + 32];
    v2f b3; b3.x = bp[48]; b3.y = bp[HID + 48];

    c0 = __builtin_amdgcn_wmma_f32_16x16x4_f32(false, a, false, b0, (short)0, c0, false, false);
    c1 = __builtin_amdgcn_wmma_f32_16x16x4_f32(false, a, false, b1, (short)0, c1, false, false);
    c2 = __builtin_amdgcn_wmma_f32_16x16x4_f32(false, a, false, b2, (short)0, c2, false, false);
    c3 = __builtin_amdgcn_wmma_f32_16x16x4_f32(false, a, false, b3, (short)0, c3, false, false);
  }

  // C/D layout: VGPR r holds M = r + 8*half, N = lane16 (per 16-col tile)
#pragma unroll
  for (int r = 0; r < 8; ++r) {
    const int row = m0 + r + 8 * half;
    if (row < nNodes) {
      float* o = h0 + (long)row * HID + lane16;
      o[0]  = c0[r];
      o[16] = c1[r];
      o[32] = c2[r];
      o[48] = c3[r];
    }
  }
}

// ---------------- layer-1 aggregation ----------------

// acc[i][:] = dinv[i]^2 * h0[i][:]   (self-loop term), float4 granularity
__global__ void k_agg1_init(const float* __restrict__ h0, const float* __restrict__ dinv,
                            float* __restrict__ acc, int n) {
  int idx = blockIdx.x * blockDim.x + threadIdx.x;   // n * 16 float4's
  if (idx < n * 16) {
    int node = idx >> 4;
    float w = dinv[node]; w *= w;
    float4 v = ((const float4*)h0)[idx];
    float4 o; o.x = v.x * w; o.y = v.y * w; o.z = v.z * w; o.w = v.w * w;
    ((float4*)acc)[idx] = o;
  }
}

// per (edge, 4-col group): acc[dst] += dinv[s]*dinv[d]*h0[src]
__global__ void k_agg1_edge(const float* __restrict__ h0, const float* __restrict__ dinv,
                            const int* __restrict__ src, const int* __restrict__ dst,
                            float* __restrict__ acc, int e) {
  int gid = blockIdx.x * blockDim.x + threadIdx.x;   // e * 16
  if (gid < e * 16) {
    int edge = gid >> 4;
    int j    = (gid & 15) << 2;
    int s = src[edge], d = dst[edge];
    float w = dinv[s] * dinv[d];
    float4 v = *(const float4*)(h0 + (long)s * HID + j);
    float* o = acc + (long)d * HID + j;
    gatomic_add_f32(o + 0, v.x * w);
    gatomic_add_f32(o + 1, v.y * w);
    gatomic_add_f32(o + 2, v.z * w);
    gatomic_add_f32(o + 3, v.w * w);
  }
}

// ---------------- layer 2: h1 = relu(acc+b1); pre = h1 @ W2 ----------------
__global__ void k_l2pre(const float* __restrict__ acc, const float* __restrict__ b1,
                        const float* __restrict__ W2, float* __restrict__ pre, int n) {
  int i = blockIdx.x * blockDim.x + threadIdx.x;
  if (i < n) {
    const float* row = acc + (long)i * HID;
    float s0 = 0.f, s1 = 0.f;
#pragma unroll
    for (int j = 0; j < HID; ++j) {
      float hv = fmaxf(row[j] + b1[j], 0.0f);
      s0 = fmaf(hv, W2[2 * j + 0], s0);
      s1 = fmaf(hv, W2[2 * j + 1], s1);
    }
    pre[2 * i + 0] = s0;
    pre[2 * i + 1] = s1;
  }
}

// out[i][c] = dinv[i]^2 * pre[i][c] + b2[c]   (self loop + bias; also un-poisons d_out)
__global__ void k_out_init(const float* __restrict__ pre, const float* __restrict__ dinv,
                           const float* __restrict__ b2, float* __restrict__ out, int n) {
  int i = blockIdx.x * blockDim.x + threadIdx.x;
  if (i < n) {
    float w = dinv[i]; w *= w;
    out[2 * i + 0] = w * pre[2 * i + 0] + b2[0];
    out[2 * i + 1] = w * pre[2 * i + 1] + b2[1];
  }
}

__global__ void k_out_edge(const float* __restrict__ pre, const float* __restrict__ dinv,
                           const int* __restrict__ src, const int* __restrict__ dst,
                           float* __restrict__ out, int e) {
  int i = blockIdx.x * blockDim.x + threadIdx.x;
  if (i < e) {
    int s = src[i], d = dst[i];
    float w = dinv[s] * dinv[d];
    gatomic_add_f32(out + 2 * d + 0, w * pre[2 * s + 0]);
    gatomic_add_f32(out + 2 * d + 1, w * pre[2 * s + 1]);
  }
}

// ---------------- launcher ----------------
extern "C" void kernel_launch(void* const* d_in, const int* in_sizes, int n_in,
                              void* d_out, int out_size, void* d_ws, size_t ws_size,
                              hipStream_t stream) {
  const float* x  = (const float*)d_in[0];
  const float* fi = (const float*)d_in[1];
  const float* W1 = (const float*)d_in[2];
  const float* b1 = (const float*)d_in[3];
  const float* W2 = (const float*)d_in[4];
  const float* b2 = (const float*)d_in[5];
  const int*   ei = (const int*)d_in[6];

  const int n = in_sizes[0] / IN_DIM;   // 100000
  const int e = in_sizes[6] / 2;        // 1600000
  const int* src = ei;
  const int* dst = ei + e;

  float* ws = (float*)d_ws;
  size_t off = 0;
  float* dinv = ws + off; off += (size_t)((n + 3) & ~3);
  float* W1p  = ws + off; off += IN_DIM * HID;
  float* h0   = ws + off; off += (size_t)n * HID;
  float* acc  = ws + off; off += (size_t)n * HID;
  float* pre  = ws + off; off += (size_t)((2 * n + 3) & ~3);
  // total ~53 MB of workspace

  float* out = (float*)d_out;
  const int T = 256;

  k_sigw1   <<<(IN_DIM * HID + T - 1) / T, T, 0, stream>>>(fi, W1, W1p);
  k_fill1   <<<(n + T - 1) / T, T, 0, stream>>>(dinv, n);
  k_degcount<<<(e + T - 1) / T, T, 0, stream>>>(dst, dinv, e);
  k_rsqrt   <<<(n + T - 1) / T, T, 0, stream>>>(dinv, n);

  const int nTiles = (n + 15) / 16;
  k_gemm1   <<<(nTiles + 3) / 4, 128, 0, stream>>>(x, W1p, h0, n, nTiles);

  k_agg1_init<<<(n * 16 + T - 1) / T, T, 0, stream>>>(h0, dinv, acc, n);
  k_agg1_edge<<<((unsigned)(e * 16) + T - 1) / T, T, 0, stream>>>(h0, dinv, src, dst, acc, e);

  k_l2pre   <<<(n + T - 1) / T, T, 0, stream>>>(acc, b1, W2, pre, n);
  k_out_init<<<(n + T - 1) / T, T, 0, stream>>>(pre, dinv, b2, out, n);
  k_out_edge<<<(e + T - 1) / T, T, 0, stream>>>(pre, dinv, src, dst, out, e);
}